// Fusedmax_25812753449288
// MI455X (gfx1250) — compile-verified
//
#include <hip/hip_runtime.h>
#include <cstdint>

// Fusedmax: sparsemax(tv1d_prox(x, ALPHA)) per row.  R=8192 rows, N=128 cols, f32.
// One thread per row; rows staged global->LDS by the CDNA5 Tensor Data Mover with
// hardware padding (LDS row stride = 129 dwords => bank-conflict-free per-lane rows).

#define N_COLS     128
#define ROWS       32          // rows per block == threads per block (one wave32)
#define LDS_STRIDE 129         // 128 + 1 pad dword inserted by TDM pad_enable
#define ALPHA      0.2f

typedef uint32_t u32x4 __attribute__((ext_vector_type(4)));
typedef uint32_t u32x8 __attribute__((ext_vector_type(8)));

__global__ __launch_bounds__(ROWS) void fusedmax_tdm_kernel(
    const float* __restrict__ x, float* __restrict__ out, int total_rows) {
  __shared__ float tile[ROWS * LDS_STRIDE];

  const int tid  = threadIdx.x;
  const int row0 = blockIdx.x * ROWS;
  const int rcnt = (total_rows - row0) < ROWS ? (total_rows - row0) : ROWS;

  // ---------------- Stage tile with the Tensor Data Mover (async DMA) -------------
  // 2-D descriptor: tile = rcnt x 128 f32; global row stride 128 dwords; TDM inserts
  // 1 pad dword per 128 dwords into LDS (pad_interval=6 -> 128 dw, pad_amount=0 -> 1 dw),
  // i.e. LDS row stride 129 dwords. OOB rows (partial last block) read as zero.
  {
    uint64_t ga       = (uint64_t)(uintptr_t)(x + (size_t)row0 * N_COLS);
    uint32_t lds_base = (uint32_t)(uintptr_t)(&tile[0]);   // low 32b of flat addr == LDS offset
    u32x4 g0;
    g0[0] = 1u;                                            // count=1 (valid), is_restore=0
    g0[1] = lds_base;                                      // lds_addr (bytes)
    g0[2] = (uint32_t)(ga & 0xFFFFFFFFu);                  // global_addr[31:0]
    g0[3] = (uint32_t)((ga >> 32) & 0x01FFFFFFu)           // global_addr[56:32]
          | (2u << 30);                                    // type = 2 ("image")
    u32x8 g1;
    g1[0] = (2u << 16)                                     // data_size = 4 bytes
          | (1u << 20)                                     // pad_enable
          | (6u << 22);                                    // pad_interval: 128 dwords (pad_amount=0 => 1 dword)
    g1[1] = (uint32_t)N_COLS << 16;                        // tensor_dim0[15:0] = 128
    g1[2] = (uint32_t)rcnt   << 16;                        // tensor_dim1[15:0] = valid rows
    g1[3] = (uint32_t)N_COLS << 16;                        // tile_dim0 = 128
    g1[4] = (uint32_t)ROWS;                                // tile_dim1 = ROWS (tile_dim2 = 0)
    g1[5] = (uint32_t)N_COLS;                              // tensor_dim0_stride = 128 elems
    g1[6] = 0u;                                            // stride hi / dim1_stride (unused, 2-D)
    g1[7] = 0u;
    asm volatile("tensor_load_to_lds %0, %1" :: "s"(g0), "s"(g1) : "memory");
    __builtin_amdgcn_s_wait_tensorcnt(0);
  }
  __syncthreads();

  float* __restrict__ r = &tile[tid * LDS_STRIDE];
  const float lam = ALPHA;
  const bool live = (tid < rcnt);

  if (live) {
    // -------- Condat TV1D prox (taut string), in place, exact branch structure ----
    int   k = 0, k0 = 0, km = 0, kp = 0;
    float vmin = r[0] - lam, vmax = r[0] + lam;
    float umin = lam, umax = -lam;
    for (int it = 0; it < 16 * N_COLS; ++it) {
      const bool  boundary = (k == N_COLS - 1);
      const float xk1  = r[boundary ? k : k + 1];          // x[min(k+1, n-1)]
      const float um_t = umin + xk1 - vmin;
      const float ux_t = umax + xk1 - vmax;
      const bool b1 = boundary && (umin < 0.f);
      const bool b2 = boundary && (umin >= 0.f) && (umax > 0.f);
      const bool b3 = boundary && (umin >= 0.f) && (umax <= 0.f);
      const bool m1 = !boundary && (um_t < -lam);
      const bool m2 = !boundary && (um_t >= -lam) && (ux_t > lam);
      const bool m3 = !boundary && (um_t >= -lam) && (ux_t <= lam);
      const bool jneg = b1 || m1;
      const bool jpos = b2 || m2;
      const int  jk = (jneg ? km : kp) + 1;
      const int  jc = jk < 0 ? 0 : (jk > N_COLS - 1 ? N_COLS - 1 : jk);
      const float xj = r[jc];                              // read BEFORE segment writes
      if (b3) {                                            // finalize & terminate
        const float wv = vmin + umin / (float)(k - k0 + 1);
        for (int i = k0; i <= k; ++i) r[i] = wv;
        break;
      }
      if (jneg || jpos) {                                  // emit constant segment
        const int   hi = jneg ? km : kp;
        const float wv = jneg ? vmin : vmax;
        for (int i = k0; i <= hi; ++i) r[i] = wv;
      }
      const float den  = (float)(k - k0 + 2);
      const bool  cmin = m3 && (um_t >= lam);
      const bool  cmax = m3 && (ux_t <= -lam);
      const int   k_n  = m3 ? k + 1 : ((jneg || jpos) ? jk : k);
      const int   k0_n = (jneg || jpos) ? jk : k0;
      const int   km_n = (b1 || m1 || m2) ? jk : (cmin ? k + 1 : km);
      const int   kp_n = (b2 || m1 || m2) ? jk : (cmax ? k + 1 : kp);
      const float vmin_n = (b1 || m1) ? xj
                         : (m2 ? xj - 2.f * lam
                         : (cmin ? vmin + (um_t - lam) / den : vmin));
      const float vmax_n = (b2 || m2) ? xj
                         : (m1 ? xj + 2.f * lam
                         : (cmax ? vmax + (ux_t + lam) / den : vmax));
      const float umin_n = b2 ? xj - lam - vmin
                         : ((b1 || m1 || m2 || cmin) ? lam : (m3 ? um_t : umin));
      const float umax_n = b1 ? xj + lam - vmax
                         : ((b2 || m1 || m2 || cmax) ? -lam : (m3 ? ux_t : umax));
      k = k_n; k0 = k0_n; km = km_n; kp = kp_n;
      vmin = vmin_n; vmax = vmax_n; umin = umin_n; umax = umax_n;
    }

    // -------- sparsemax: exact simplex projection, sort-free (Michelot) ----------
    {
      float s = 0.f;
      for (int i = 0; i < N_COLS; ++i) s += r[i];
      float tau = (s - 1.f) / (float)N_COLS;
      int c = N_COLS;
      for (int it = 0; it < N_COLS; ++it) {
        float s2 = 0.f; int c2 = 0;
        for (int i = 0; i < N_COLS; ++i) {
          const float v = r[i];
          if (v > tau) { s2 += v; ++c2; }
        }
        if (c2 == c || c2 == 0) break;
        c = c2; tau = (s2 - 1.f) / (float)c2;
      }
      for (int i = 0; i < N_COLS; ++i) r[i] = fmaxf(r[i] - tau, 0.f);
    }
  }
  __syncthreads();

  // ---------------- Coalesced store: read LDS transposed, write linear ------------
  const int valid = rcnt * N_COLS;
  float* gout = out + (size_t)row0 * N_COLS;
  for (int kk = 0; kk < N_COLS; ++kk) {
    const int f = tid + kk * ROWS;                 // 0 .. ROWS*128-1, lane-contiguous
    if (f < valid) {
      const int rr = f >> 7;                       // f / 128
      const int cc = f & (N_COLS - 1);             // f % 128
      gout[f] = tile[rr * LDS_STRIDE + cc];        // consecutive LDS addrs: no conflicts
    }
  }
}

extern "C" void kernel_launch(void* const* d_in, const int* in_sizes, int n_in,
                              void* d_out, int out_size, void* d_ws, size_t ws_size,
                              hipStream_t stream) {
  const float* x = (const float*)d_in[0];
  float* out = (float*)d_out;
  const int total = in_sizes[0];                   // 8192 * 128
  const int rows  = total / N_COLS;                // 8192
  const int blocks = (rows + ROWS - 1) / ROWS;     // 256 one-wave blocks
  fusedmax_tdm_kernel<<<blocks, ROWS, 0, stream>>>(x, out, rows);
}